// GCNBlock_20074677142000
// MI455X (gfx1250) — compile-verified
//
#include <hip/hip_runtime.h>
#include <math.h>

typedef __attribute__((ext_vector_type(2))) float v2f;
typedef __attribute__((ext_vector_type(8))) float v8f;

__device__ __forceinline__ float sigmoidf_(float v) { return 1.0f / (1.0f + expf(-v)); }
__device__ __forceinline__ float gelu_(float v) {
  return 0.5f * v * (1.0f + erff(v * 0.70710678118654752f));
}

// ---------------- degree / norm kernels ----------------

__global__ void k_fill1(float* __restrict__ p, int n) {
  int i = blockIdx.x * blockDim.x + threadIdx.x;
  if (i < n) p[i] = 1.0f;  // self-loop weight 1 pre-added
}

__global__ void k_deg_accum(const float* __restrict__ ew, const long long* __restrict__ ei,
                            float* deg, int E) {
  int e = blockIdx.x * blockDim.x + threadIdx.x;
  if (e < E) {
    int c = (int)ei[(size_t)E + e];           // col
    atomicAdd(&deg[c], sigmoidf_(ew[e]));
  }
}

__global__ void k_dinv(float* deg, int n) {
  int i = blockIdx.x * blockDim.x + threadIdx.x;
  if (i < n) {
    float d = deg[i];
    deg[i] = (d > 0.0f) ? rsqrtf(d) : 0.0f;   // in-place deg -> dinv
  }
}

__global__ void k_norm(const float* __restrict__ ew, const long long* __restrict__ ei,
                       const float* __restrict__ dinv, float* __restrict__ nrm, int E) {
  int e = blockIdx.x * blockDim.x + threadIdx.x;
  if (e < E) {
    int r = (int)ei[e];
    int c = (int)ei[(size_t)E + e];
    nrm[e] = dinv[r] * sigmoidf_(ew[e]) * dinv[c];
  }
}

// ---------------- WMMA GEMM: H[N,64] = X[N,64] @ W^T (W row-major [64,64]) ----------------
// One wave per 16-row tile; 4 x (16x16) C tiles cover all 64 output columns.
// V_WMMA_F32_16X16X4_F32: A 16x4 f32, B 4x16 f32, C/D 16x16 f32.
// A layout: lanes 0-15 = M, vgpr{0,1} = K{0,1}; lanes 16-31 = M, K{2,3}.
// B layout (symmetric): lanes 0-15 = N, vgpr{0,1} = K{0,1}; lanes 16-31 = K{2,3}.

__global__ void k_gemm64(const float* __restrict__ X, const float* __restrict__ W,
                         float* __restrict__ H, int N, int nTiles) {
  int wid  = threadIdx.x >> 5;
  int tile = blockIdx.x * (blockDim.x >> 5) + wid;
  if (tile >= nTiles) return;                  // uniform per wave -> EXEC stays all-1s
  int lane  = threadIdx.x & 31;
  int laneM = lane & 15;
  int half  = lane >> 4;
  int r0 = tile * 16;
  int arow = r0 + laneM; if (arow >= N) arow = N - 1;   // clamp (keeps EXEC full)

  const float* xA  = X + (size_t)arow * 64 + 2 * half;
  const float* wB0 = W + (size_t)( 0 + laneM) * 64 + 2 * half;
  const float* wB1 = W + (size_t)(16 + laneM) * 64 + 2 * half;
  const float* wB2 = W + (size_t)(32 + laneM) * 64 + 2 * half;
  const float* wB3 = W + (size_t)(48 + laneM) * 64 + 2 * half;

  v8f c0 = {}, c1 = {}, c2 = {}, c3 = {};
#pragma unroll
  for (int k0 = 0; k0 < 64; k0 += 4) {
    v2f a  = *(const v2f*)(xA  + k0);
    v2f b0 = *(const v2f*)(wB0 + k0);
    v2f b1 = *(const v2f*)(wB1 + k0);
    v2f b2 = *(const v2f*)(wB2 + k0);
    v2f b3 = *(const v2f*)(wB3 + k0);
    c0 = __builtin_amdgcn_wmma_f32_16x16x4_f32(false, a, false, b0, (short)0, c0, false, false);
    c1 = __builtin_amdgcn_wmma_f32_16x16x4_f32(false, a, false, b1, (short)0, c1, false, false);
    c2 = __builtin_amdgcn_wmma_f32_16x16x4_f32(false, a, false, b2, (short)0, c2, false, false);
    c3 = __builtin_amdgcn_wmma_f32_16x16x4_f32(false, a, false, b3, (short)0, c3, false, false);
  }

  // C/D layout: lanes 0-15 -> N=lane, M=v; lanes 16-31 -> M=v+8
#pragma unroll
  for (int v = 0; v < 8; ++v) {
    int row = r0 + v + 8 * half;
    if (row < N) {
      float* hr = H + (size_t)row * 64 + laneM;
      hr[0]  = c0[v];
      hr[16] = c1[v];
      hr[32] = c2[v];
      hr[48] = c3[v];
    }
  }
}

// ---------------- aggregation ----------------

// agg[i,c] = bias[c] + h[i,c] * dinv[i]^2   (bias + self-loop contribution)
__global__ void k_agg_init(const float* __restrict__ h, const float* __restrict__ dinv,
                           const float* __restrict__ bias, float* __restrict__ agg, int total) {
  int i = blockIdx.x * blockDim.x + threadIdx.x;
  if (i < total) {
    int node = i >> 6;
    int c    = i & 63;
    float di = dinv[node];
    agg[i] = bias[c] + h[i] * di * di;
  }
}

// agg[col] += h[row] * norm[e], 16 threads/edge, float4 per thread
__global__ void k_scatter(const float* __restrict__ h, const long long* __restrict__ ei,
                          const float* __restrict__ nrm, float* agg, int E) {
  long long idx = (long long)blockIdx.x * blockDim.x + threadIdx.x;
  if (idx < (long long)E * 16) {
    int e = (int)(idx >> 4);
    int q = (int)(idx & 15);
    int r = (int)ei[e];
    int c = (int)ei[(size_t)E + e];
    float nv = nrm[e];
    float4 hv = ((const float4*)h)[(size_t)r * 16 + q];
    float* dst = agg + (size_t)c * 64 + (size_t)q * 4;
    atomicAdd(dst + 0, hv.x * nv);
    atomicAdd(dst + 1, hv.y * nv);
    atomicAdd(dst + 2, hv.z * nv);
    atomicAdd(dst + 3, hv.w * nv);
  }
}

// ---------------- layernorm (+gelu) : one wave32 per row, 2 channels/lane ----------------

__global__ void k_ln_gelu(const float* __restrict__ in, const float* __restrict__ g,
                          const float* __restrict__ be, float* __restrict__ out, int nrows) {
  int wid = threadIdx.x >> 5;
  int row = blockIdx.x * (blockDim.x >> 5) + wid;
  if (row >= nrows) return;
  int lane = threadIdx.x & 31;
  const float* p = in + (size_t)row * 64;
  float a0 = p[lane], a1 = p[lane + 32];
  float s = a0 + a1;
#pragma unroll
  for (int off = 16; off > 0; off >>= 1) s += __shfl_xor(s, off, 32);
  float mu = s * (1.0f / 64.0f);
  float d0 = a0 - mu, d1 = a1 - mu;
  float v = d0 * d0 + d1 * d1;
#pragma unroll
  for (int off = 16; off > 0; off >>= 1) v += __shfl_xor(v, off, 32);
  float rs = rsqrtf(v * (1.0f / 64.0f) + 1e-6f);
  float* q = out + (size_t)row * 64;
  q[lane]      = gelu_(d0 * rs * g[lane]      + be[lane]);
  q[lane + 32] = gelu_(d1 * rs * g[lane + 32] + be[lane + 32]);
}

// out = gelu(layernorm(in)*g + be + x)   (in may alias out; each lane reads before writing)
__global__ void k_ln_res_gelu(const float* in, const float* __restrict__ x,
                              const float* __restrict__ g, const float* __restrict__ be,
                              float* out, int nrows) {
  int wid = threadIdx.x >> 5;
  int row = blockIdx.x * (blockDim.x >> 5) + wid;
  if (row >= nrows) return;
  int lane = threadIdx.x & 31;
  const float* p  = in + (size_t)row * 64;
  const float* xr = x  + (size_t)row * 64;
  float a0 = p[lane], a1 = p[lane + 32];
  float s = a0 + a1;
#pragma unroll
  for (int off = 16; off > 0; off >>= 1) s += __shfl_xor(s, off, 32);
  float mu = s * (1.0f / 64.0f);
  float d0 = a0 - mu, d1 = a1 - mu;
  float v = d0 * d0 + d1 * d1;
#pragma unroll
  for (int off = 16; off > 0; off >>= 1) v += __shfl_xor(v, off, 32);
  float rs = rsqrtf(v * (1.0f / 64.0f) + 1e-6f);
  float r0 = gelu_(d0 * rs * g[lane]      + be[lane]      + xr[lane]);
  float r1 = gelu_(d1 * rs * g[lane + 32] + be[lane + 32] + xr[lane + 32]);
  float* q = out + (size_t)row * 64;
  q[lane]      = r0;
  q[lane + 32] = r1;
}

// ---------------- launcher ----------------

extern "C" void kernel_launch(void* const* d_in, const int* in_sizes, int n_in,
                              void* d_out, int out_size, void* d_ws, size_t ws_size,
                              hipStream_t stream) {
  const float*     x   = (const float*)d_in[0];
  const long long* ei  = (const long long*)d_in[1];   // int64 [2,E]
  const float*     ew  = (const float*)d_in[2];
  const float*     W1  = (const float*)d_in[3];
  const float*     b1  = (const float*)d_in[4];
  const float*     g1  = (const float*)d_in[5];
  const float*     be1 = (const float*)d_in[6];
  const float*     W2  = (const float*)d_in[7];
  const float*     b2  = (const float*)d_in[8];
  const float*     g2  = (const float*)d_in[9];
  const float*     be2 = (const float*)d_in[10];

  const int N = in_sizes[0] / 64;
  const int E = in_sizes[2];

  float* ws   = (float*)d_ws;
  float* dinv = ws;                       // N floats (deg then dinv, in place)
  float* nrm  = dinv + N;                 // E floats
  float* h    = nrm + E;                  // N*64 floats
  float* h2   = h + (size_t)N * 64;       // N*64 floats
  float* out  = (float*)d_out;

  const int TB = 256;
  const int nTiles = (N + 15) / 16;
  const int gemmBlocks = (nTiles + 3) / 4;               // 128 thr = 4 waves/block
  const int lnBlocks   = (N + 3) / 4;                    // 128 thr = 4 rows/block
  const long long scat = (long long)E * 16;
  const int scatBlocks = (int)((scat + TB - 1) / TB);
  const int totNC = N * 64;

  // normalization coefficients (shared by both convs)
  k_fill1    <<<(N + TB - 1) / TB, TB, 0, stream>>>(dinv, N);
  k_deg_accum<<<(E + TB - 1) / TB, TB, 0, stream>>>(ew, ei, dinv, E);
  k_dinv     <<<(N + TB - 1) / TB, TB, 0, stream>>>(dinv, N);
  k_norm     <<<(E + TB - 1) / TB, TB, 0, stream>>>(ew, ei, dinv, nrm, E);

  // conv1: h = x @ W1^T ; h2 = b1 + selfloop + scatter ; h = gelu(ln(h2))
  k_gemm64   <<<gemmBlocks, 128, 0, stream>>>(x, W1, h, N, nTiles);
  k_agg_init <<<(totNC + TB - 1) / TB, TB, 0, stream>>>(h, dinv, b1, h2, totNC);
  k_scatter  <<<scatBlocks, TB, 0, stream>>>(h, ei, nrm, h2, E);
  k_ln_gelu  <<<lnBlocks, 128, 0, stream>>>(h2, g1, be1, h, N);

  // conv2: h2 = h @ W2^T ; out = b2 + selfloop + scatter ; out = gelu(ln(out) + x)
  k_gemm64   <<<gemmBlocks, 128, 0, stream>>>(h, W2, h2, N, nTiles);
  k_agg_init <<<(totNC + TB - 1) / TB, TB, 0, stream>>>(h2, dinv, b2, out, totNC);
  k_scatter  <<<scatBlocks, TB, 0, stream>>>(h2, ei, nrm, out, E);
  k_ln_res_gelu<<<lnBlocks, 128, 0, stream>>>(out, x, g2, be2, out, N);

  (void)n_in; (void)out_size; (void)ws_size;
}